// IOULoss2d_19516331393757
// MI455X (gfx1250) — compile-verified
//
#include <hip/hip_runtime.h>

// ---------------------------------------------------------------------------
// IoU loss: inputs [8,21,512,512] f32 logits, targets [8,512,512] int32.
// Per pixel: online softmax over C=21 (one streaming pass), sparse per-class
// accumulation (only amax / target classes contribute), then tiny finalize.
// Memory-bound: ~185 MB single-pass -> ~8 us floor at 23.3 TB/s.
// ---------------------------------------------------------------------------

typedef __attribute__((ext_vector_type(4))) float v4f;
typedef __attribute__((ext_vector_type(4))) int   v4i;
typedef __attribute__((ext_vector_type(2))) float v2f;
typedef __attribute__((ext_vector_type(8))) float v8f;

static constexpr int NB = 8;
static constexpr int NC = 21;
static constexpr int HW = 512 * 512;

// ---------------------------------------------------------------------------
// Main streaming kernel: each thread handles 4 consecutive pixels (b128 NT
// loads, fully coalesced: 32 lanes x 16B = 512B per wave per class-row).
// Online softmax: running max m, rescaled sum s, argmax, and x[target].
// Sparse accumulation into 21+21 LDS float buckets, then ~42 global atomics.
// ---------------------------------------------------------------------------
__global__ __launch_bounds__(256) void iou_partial_kernel(
    const float* __restrict__ logits,
    const int*   __restrict__ targets,
    float*       __restrict__ g_acc)   // [2][NB][NC]: union then intersect
{
    __shared__ float s_u[NC];
    __shared__ float s_i[NC];

    const int tid = threadIdx.x;
    if (tid < NC) { s_u[tid] = 0.0f; s_i[tid] = 0.0f; }
    __syncthreads();

    const int b = blockIdx.y;
    const int g = blockIdx.x * blockDim.x + tid;   // pixel-quad index
    const int p = g * 4;                           // first pixel of quad

    const float* base = logits + ((size_t)b * NC) * HW + p;

    // Target labels for 4 pixels (16B NT load).
    v4i t4 = __builtin_nontemporal_load(
        (const v4i*)(targets + (size_t)b * HW + p));
    int tcl[4] = { t4.x, t4.y, t4.z, t4.w };

    // Online softmax state per pixel.
    v4f x0 = __builtin_nontemporal_load((const v4f*)base);
    float xv0[4] = { x0.x, x0.y, x0.z, x0.w };
    float m[4], s[4], xt[4];
    int amax[4];
#pragma unroll
    for (int j = 0; j < 4; ++j) {
        m[j] = xv0[j]; s[j] = 1.0f; amax[j] = 0;
        xt[j] = (tcl[j] == 0) ? xv0[j] : 0.0f;
    }

#pragma unroll
    for (int c = 1; c < NC; ++c) {
        v4f x = __builtin_nontemporal_load((const v4f*)(base + (size_t)c * HW));
        float xc[4] = { x.x, x.y, x.z, x.w };
#pragma unroll
        for (int j = 0; j < 4; ++j) {
            if (tcl[j] == c) xt[j] = xc[j];
            if (xc[j] > m[j]) {                    // strict > keeps first-max
                s[j] = s[j] * __expf(m[j] - xc[j]) + 1.0f;
                m[j] = xc[j];
                amax[j] = c;
            } else {
                s[j] += __expf(xc[j] - m[j]);
            }
        }
    }

#pragma unroll
    for (int j = 0; j < 4; ++j) {
        float inv = 1.0f / s[j];                   // prob of argmax class
        atomicAdd(&s_u[amax[j]], inv);             // union @ amax (always)
        if (tcl[j] == amax[j]) {
            atomicAdd(&s_i[tcl[j]], inv);          // intersect @ t == amax
        } else {
            atomicAdd(&s_u[tcl[j]], __expf(xt[j] - m[j]) * inv); // union @ t
        }
    }

    __syncthreads();
    if (tid < NC) {
        atomicAdd(&g_acc[b * NC + tid],           s_u[tid]);
        atomicAdd(&g_acc[NB * NC + b * NC + tid], s_i[tid]);
    }
}

// ---------------------------------------------------------------------------
// Finalize (one wave32): ratio = intersect / max(union,1); val = (ratio-1)^2/B
// Total-sum of the 168 values via chained V_WMMA_F32_16X16X4_F32 with an
// all-ones A: D[m][n] = sum_k B[k][n], so summing lane-VGPR0 over all 32
// lanes yields 2x the grand total regardless of B's exact fragment layout.
// ---------------------------------------------------------------------------
__global__ __launch_bounds__(32) void iou_finalize_kernel(
    const float* __restrict__ g_acc,
    float*       __restrict__ out)
{
    const int lane = threadIdx.x;                  // 0..31, wave32

    v2f ones; ones.x = 1.0f; ones.y = 1.0f;        // A = ones(16x4)
    v8f acc = {};                                  // C/D accumulator (16x16)

#pragma unroll
    for (int t = 0; t < 4; ++t) {
        v2f bfrag;
#pragma unroll
        for (int comp = 0; comp < 2; ++comp) {
            int idx = t * 64 + comp * 32 + lane;   // 0..255 slot
            int idc = idx < NB * NC ? idx : 0;     // clamped, branchless
            float u  = g_acc[idc];
            float iv = g_acc[NB * NC + idc];
            float ratio = iv / fmaxf(u, 1.0f);
            float d = ratio - 1.0f;
            float val = d * d * (1.0f / NB);
            bfrag[comp] = (idx < NB * NC) ? val : 0.0f;
        }
        // 8 args: (neg_a, A, neg_b, B, c_mod, C, reuse_a, reuse_b)
        acc = __builtin_amdgcn_wmma_f32_16x16x4_f32(
            false, ones, false, bfrag, (short)0, acc, false, false);
    }

    // VGPR0 of D holds column sums in every lane (rows 0 and 8 identical):
    // sum over 32 lanes = 2 * grand total.
    float e = acc[0];
#pragma unroll
    for (int off = 16; off > 0; off >>= 1)
        e += __shfl_xor(e, off, 32);

    if (lane == 0) out[0] = e * 0.5f;
}

// ---------------------------------------------------------------------------
extern "C" void kernel_launch(void* const* d_in, const int* in_sizes, int n_in,
                              void* d_out, int out_size, void* d_ws, size_t ws_size,
                              hipStream_t stream)
{
    const float* logits  = (const float*)d_in[0];
    const int*   targets = (const int*)d_in[1];
    float*       out     = (float*)d_out;
    float*       acc     = (float*)d_ws;          // 2*NB*NC floats = 1344 B

    (void)in_sizes; (void)n_in; (void)out_size; (void)ws_size;

    hipMemsetAsync(acc, 0, 2 * NB * NC * sizeof(float), stream);

    // HW/4 pixel-quads per image = 65536 = 256 blocks x 256 threads.
    dim3 grid(HW / 4 / 256, NB, 1);
    iou_partial_kernel<<<grid, 256, 0, stream>>>(logits, targets, acc);

    iou_finalize_kernel<<<1, 32, 0, stream>>>(acc, out);
}